// RecurrentCPPN_53893249630523
// MI455X (gfx1250) — compile-verified
//
#include <hip/hip_runtime.h>

typedef __attribute__((ext_vector_type(2))) float v2f;
typedef __attribute__((ext_vector_type(8))) float v8f;

#define NDATA   262144
#define NIN     4
#define MSTATE  64
#define DEG     6
#define EDGES   (MSTATE * DEG)      // 384
#define KTOT    (NIN + MSTATE)      // 68 = all_act width
#define KSTEPS  (KTOT / 4)          // 17 K-steps of 4 for f32 WMMA
#define THREADS 256
#define ROWS_PER_BLOCK 128          // 8 waves * 16 rows

// One recurrent CPPN step:
//   new_state = act( bias + response * (all_act @ W) ),  all_act = [input | prev_state]
// W[68][64] is densified from the edge list and stored in LDS directly in the
// V_WMMA_F32_16X16X4_F32 B-fragment register layout:
//   lane L: col N = L%16 ; vgpr j holds K = 2*(L/16) + j   (K within a 4-step)
__global__ __launch_bounds__(THREADS)
void RecurrentCPPN_step_kernel(const float* __restrict__ input,       // [N,4]
                               const float* __restrict__ prev_state,  // [N,64]
                               const float* __restrict__ edge_w,      // [E]
                               const float* __restrict__ bias,        // [M]
                               const float* __restrict__ response,    // [M]
                               const int*   __restrict__ edge_src,    // [E]
                               const int*   __restrict__ act_id,      // [M]
                               float* __restrict__ out_head,          // [N,4]
                               float* __restrict__ out_state)         // [N,64]
{
    // B fragments: [KSTEPS][4 neuron-tiles][32 lanes] of float2
    __shared__ v2f   sB[KSTEPS * 4 * 32];
    __shared__ float sBias[MSTATE];
    __shared__ float sResp[MSTATE];
    __shared__ int   sAct[MSTATE];

    float* fB = (float*)sB;

    // ---- build dense weight fragments in LDS (deterministic every call) ----
    for (int i = threadIdx.x; i < KSTEPS * 4 * 32 * 2; i += THREADS) fB[i] = 0.0f;
    if (threadIdx.x < MSTATE) {
        sBias[threadIdx.x] = bias[threadIdx.x];
        sResp[threadIdx.x] = response[threadIdx.x];
        sAct[threadIdx.x]  = act_id[threadIdx.x];
    }
    __syncthreads();

    for (int e = threadIdx.x; e < EDGES; e += THREADS) {
        int   src  = edge_src[e];          // 0..67, row of W (K index)
        float w    = edge_w[e];
        int   dst  = e / DEG;              // 0..63, col of W (neuron)
        int   k    = src >> 2;             // K-step
        int   koff = src & 3;              // K within step
        int   lane = ((koff >> 1) << 4) | (dst & 15);
        int   idx  = ((((k << 2) + (dst >> 4)) * 32 + lane) << 1) | (koff & 1);
        atomicAdd(&fB[idx], w);            // duplicates (src,dst) accumulate
    }
    __syncthreads();

    // ---- per-wave 16x64 output slab ----
    const int lane = threadIdx.x & 31;
    const int wave = threadIdx.x >> 5;
    const int grp  = lane >> 4;            // lane group (K split for A/B frags)
    const int r    = lane & 15;            // row (A) / col (B,D) within tile
    const int base = blockIdx.x * ROWS_PER_BLOCK + wave * 16;

    const float* inP = input      + (size_t)(base + r) * NIN;
    const float* psP = prev_state + (size_t)(base + r) * MSTATE;

    v8f acc[4] = {};   // 4 neuron tiles of 16

    // K-step 0: columns 0..3 of all_act come from `input`
    {
        v2f a = *(const v2f*)(inP + grp * 2);
#pragma unroll
        for (int t = 0; t < 4; ++t) {
            v2f b = sB[t * 32 + lane];
            acc[t] = __builtin_amdgcn_wmma_f32_16x16x4_f32(
                false, a, false, b, (short)0, acc[t], false, false);
        }
    }
    // K-steps 1..16: columns 4k..4k+3 come from prev_state cols 4(k-1)..
#pragma unroll
    for (int k = 1; k < KSTEPS; ++k) {
        v2f a = *(const v2f*)(psP + (k - 1) * 4 + grp * 2);
#pragma unroll
        for (int t = 0; t < 4; ++t) {
            v2f b = sB[(k * 4 + t) * 32 + lane];
            acc[t] = __builtin_amdgcn_wmma_f32_16x16x4_f32(
                false, a, false, b, (short)0, acc[t], false, false);
        }
    }

    // ---- epilogue: bias/response + per-neuron activation, store ----
#pragma unroll
    for (int t = 0; t < 4; ++t) {
        const int   m   = t * 16 + r;      // neuron index for this lane
        const float bi  = sBias[m];
        const float re  = sResp[m];
        const int   aid = sAct[m];
#pragma unroll
        for (int j = 0; j < 8; ++j) {
            const int   n = base + 8 * grp + j;   // data row for D vgpr j
            const float p = bi + re * acc[t][j];
            float y;
            if (aid == 0)      y = 1.0f / (1.0f + __expf(-p));            // sigmoid
            else if (aid == 1) y = 2.0f / (1.0f + __expf(-2.0f * p)) - 1.0f; // tanh
            else if (aid == 2) y = __sinf(p);
            else               y = fabsf(p);
            out_state[(size_t)n * MSTATE + m] = y;
            if (m >= MSTATE - NIN)
                out_head[(size_t)n * NIN + (m - (MSTATE - NIN))] = y;
        }
    }
}

extern "C" void kernel_launch(void* const* d_in, const int* in_sizes, int n_in,
                              void* d_out, int out_size, void* d_ws, size_t ws_size,
                              hipStream_t stream) {
    (void)in_sizes; (void)n_in; (void)d_ws; (void)ws_size; (void)out_size;
    const float* input      = (const float*)d_in[0];
    const float* prev_state = (const float*)d_in[1];
    const float* edge_w     = (const float*)d_in[2];
    const float* bias       = (const float*)d_in[3];
    const float* response   = (const float*)d_in[4];
    const int*   edge_src   = (const int*)  d_in[5];
    // d_in[6] = edge_dst: deterministic repeat(arange(64),6), derived as e/DEG
    const int*   act_id     = (const int*)  d_in[7];

    float* out_head  = (float*)d_out;                       // [N,4]
    float* out_state = (float*)d_out + (size_t)NDATA * NIN; // [N,64]

    dim3 grid(NDATA / ROWS_PER_BLOCK);
    dim3 block(THREADS);
    hipLaunchKernelGGL(RecurrentCPPN_step_kernel, grid, block, 0, stream,
                       input, prev_state, edge_w, bias, response,
                       edge_src, act_id, out_head, out_state);
}